// EdgeConv_31387620999471
// MI455X (gfx1250) — compile-verified
//
#include <hip/hip_runtime.h>

typedef float v2f __attribute__((ext_vector_type(2)));
typedef float v8f __attribute__((ext_vector_type(8)));

#define BATCH 4
#define NPTS  8192
#define NCH   32
#define KNN   20
#define OC    64
#define COLT  128
#define CNT_ELEMS (BATCH * NPTS * KNN)   // 655360 elements per channel for BN

static __device__ __forceinline__ v8f wmma_f32(v2f a, v2f b, v8f c) {
    // V_WMMA_F32_16X16X4_F32: D = A(16x4) * B(4x16) + C
    return __builtin_amdgcn_wmma_f32_16x16x4_f32(
        /*neg_a=*/false, a, /*neg_b=*/false, b,
        /*c_mod=*/(short)0, c, /*reuse_a=*/false, /*reuse_b=*/false);
}

// ---------------------------------------------------------------------------
// Kernel 0: build combined weight Wc (128x32) and zero BN stats accumulators.
//   Wc[o][c]      = W0[o][c]                      (o <  64)  -> y1 = W0a @ x
//   Wc[64+o][c]   = W0[o][32+c] - W0[o][c]        (o >= 64)  -> y2 = (W0b-W0a) @ x
// ---------------------------------------------------------------------------
__global__ void ec_prep(const float* __restrict__ W0, float* __restrict__ Wc,
                        float* __restrict__ stats) {
    int t = blockIdx.x * blockDim.x + threadIdx.x;
    for (int i = t; i < 128 * NCH; i += gridDim.x * blockDim.x) {
        int o = i >> 5, c = i & 31;
        float w = (o < OC) ? W0[o * 64 + c]
                           : (W0[(o - OC) * 64 + 32 + c] - W0[(o - OC) * 64 + c]);
        Wc[i] = w;
    }
    if (t < 128) stats[t] = 0.0f;   // 64 sums + 64 sums-of-squares
}

// ---------------------------------------------------------------------------
// Kernel 1: fused WMMA distance tiles + streaming top-20 selection.
// Block = 128 threads (4 waves). Each wave owns 32 rows; block owns 128 rows.
// Column tiles of 128 points staged in LDS; 16x16 dist tiles via 8+8 WMMAs.
// ---------------------------------------------------------------------------
__global__ __launch_bounds__(128) void ec_knn(const float* __restrict__ xg,
                                              int* __restrict__ idxb) {
    __shared__ float xcol[NCH][COLT];   // 16 KB column tile (channel-major)
    __shared__ float sqm[COLT];         // squared norms of column points
    __shared__ float dt[4][32][17];     // per-wave 32x16 Gram scratch (padded)

    const int lane = threadIdx.x & 31;
    const int wave = threadIdx.x >> 5;
    const int b    = blockIdx.x >> 6;            // 8192/128 = 64 row-blocks/batch
    const int rb0  = (blockIdx.x & 63) << 7;
    const int rowBase = rb0 + wave * 32;
    const int mrow = lane & 15;                  // M (or N) index within tile
    const int kp   = lane >> 4;                  // K pair selector (ISA layout)
    const float* xb = xg + b * NCH * NPTS;

    // Preload A operands: 32 rows x 32 channels, ISA 16x4 f32 A layout.
    v2f a0[8], a1[8];
#pragma unroll
    for (int kb = 0; kb < 8; ++kb) {
        int c0 = 4 * kb + 2 * kp;
        a0[kb].x = xb[c0 * NPTS + rowBase + mrow];
        a0[kb].y = xb[(c0 + 1) * NPTS + rowBase + mrow];
        a1[kb].x = xb[c0 * NPTS + rowBase + 16 + mrow];
        a1[kb].y = xb[(c0 + 1) * NPTS + rowBase + 16 + mrow];
    }

    const int myRow = rowBase + lane;            // selection row owned by lane
    float sqr = 0.0f;
    for (int c = 0; c < NCH; ++c) { float v = xb[c * NPTS + myRow]; sqr += v * v; }

    // Unsorted top-20 (smallest distance) kept fully in registers.
    float topd[KNN]; int topi[KNN];
#pragma unroll
    for (int j = 0; j < KNN; ++j) { topd[j] = 3.0e38f; topi[j] = 0; }
    float worst = 3.0e38f; int worstPos = 0;

    for (int m0 = 0; m0 < NPTS; m0 += COLT) {
        __syncthreads();                          // WAR on xcol/sqm
        {
            int t = threadIdx.x; float s = 0.0f;
            for (int c = 0; c < NCH; ++c) {
                float v = xb[c * NPTS + m0 + t];
                xcol[c][t] = v; s += v * v;
            }
            sqm[t] = s;
        }
        if (m0 + COLT < NPTS)                     // global_prefetch_b8 of next tile
            __builtin_prefetch(xb + m0 + COLT + threadIdx.x, 0, 1);
        __syncthreads();

        for (int mt = 0; mt < COLT; mt += 16) {
            v8f acc0 = {}, acc1 = {};
#pragma unroll
            for (int kb = 0; kb < 8; ++kb) {
                int c0 = 4 * kb + 2 * kp;
                v2f bb;
                bb.x = xcol[c0][mt + mrow];
                bb.y = xcol[c0 + 1][mt + mrow];
                acc0 = wmma_f32(a0[kb], bb, acc0);
                acc1 = wmma_f32(a1[kb], bb, acc1);
            }
            // Scatter D (ISA C/D layout) into wave-private LDS scratch.
#pragma unroll
            for (int r = 0; r < 8; ++r) {
                dt[wave][r + 8 * kp][mrow]      = acc0[r];
                dt[wave][16 + r + 8 * kp][mrow] = acc1[r];
            }
            __builtin_amdgcn_wave_barrier();
            asm volatile("" ::: "memory");

            // Each lane scans its row's 16 candidates.
            for (int j = 0; j < 16; ++j) {
                int m = m0 + mt + j;
                float g = dt[wave][lane][j];
                float d = sqr + sqm[mt + j] - 2.0f * g;
                if (m != myRow && d < worst) {
#pragma unroll
                    for (int t = 0; t < KNN; ++t)
                        if (t == worstPos) { topd[t] = d; topi[t] = m; }
                    worst = -3.0e38f;
#pragma unroll
                    for (int t = 0; t < KNN; ++t)
                        if (topd[t] > worst) { worst = topd[t]; worstPos = t; }
                }
            }
            __builtin_amdgcn_wave_barrier();
            asm volatile("" ::: "memory");
        }
    }
    int* op = idxb + (b * NPTS + myRow) * KNN;
#pragma unroll
    for (int j = 0; j < KNN; ++j) op[j] = topi[j];
}

// ---------------------------------------------------------------------------
// Kernel 2: y(128 x 8192) = Wc(128x32) @ x(32x8192) per batch, via WMMA.
// Block = 128 threads (4 waves); each wave does 32 o-rows x 16 n-cols.
// ---------------------------------------------------------------------------
__global__ __launch_bounds__(128) void ec_gemm(const float* __restrict__ xg,
                                               const float* __restrict__ Wc,
                                               float* __restrict__ y) {
    const int lane = threadIdx.x & 31;
    const int wave = threadIdx.x >> 5;
    const int b     = blockIdx.x >> 9;           // 512 n-tiles per batch
    const int ncol0 = (blockIdx.x & 511) << 4;
    const int mrow = lane & 15;
    const int kp   = lane >> 4;
    const float* xb = xg + b * NCH * NPTS;

    v2f bb[8];
#pragma unroll
    for (int kb = 0; kb < 8; ++kb) {
        int c0 = 4 * kb + 2 * kp;
        bb[kb].x = xb[c0 * NPTS + ncol0 + mrow];
        bb[kb].y = xb[(c0 + 1) * NPTS + ncol0 + mrow];
    }
    const int obase = wave * 32;
#pragma unroll
    for (int h = 0; h < 2; ++h) {
        v2f aw[8];
        int orow = obase + 16 * h + mrow;
#pragma unroll
        for (int kb = 0; kb < 8; ++kb) {
            int c0 = 4 * kb + 2 * kp;
            aw[kb].x = Wc[orow * NCH + c0];
            aw[kb].y = Wc[orow * NCH + c0 + 1];
        }
        v8f acc = {};
#pragma unroll
        for (int kb = 0; kb < 8; ++kb) acc = wmma_f32(aw[kb], bb[kb], acc);
#pragma unroll
        for (int r = 0; r < 8; ++r) {
            int o = obase + 16 * h + r + 8 * kp;
            y[(b * 128 + o) * NPTS + ncol0 + mrow] = acc[r];
        }
    }
}

// ---------------------------------------------------------------------------
// Kernel 3: BN statistics. One block per (b, o); y1/y2 rows staged in LDS.
// h[b,o,n,k] = y1[o, idx[n,k]] + y2[o, n]; accumulate sum and sum^2.
// ---------------------------------------------------------------------------
__global__ __launch_bounds__(256) void ec_stats(const float* __restrict__ y,
                                                const int* __restrict__ idxb,
                                                float* __restrict__ stats) {
    __shared__ float y1s[NPTS];   // 32 KB
    __shared__ float y2s[NPTS];   // 32 KB
    const int b = blockIdx.x >> 6;
    const int o = blockIdx.x & 63;
    const float* y1 = y + (b * 128 + o) * NPTS;
    const float* y2 = y + (b * 128 + OC + o) * NPTS;
    for (int n = threadIdx.x; n < NPTS; n += blockDim.x) {
        y1s[n] = y1[n]; y2s[n] = y2[n];
    }
    __syncthreads();

    float s = 0.0f, ss = 0.0f;
    for (int n = threadIdx.x; n < NPTS; n += blockDim.x) {
        float base = y2s[n];
        const int* ir = idxb + (b * NPTS + n) * KNN;
#pragma unroll
        for (int k = 0; k < KNN; ++k) {
            float v = y1s[ir[k]] + base;
            s += v; ss += v * v;
        }
    }
#pragma unroll
    for (int off = 16; off > 0; off >>= 1) {
        s  += __shfl_down(s,  off, 32);
        ss += __shfl_down(ss, off, 32);
    }
    if ((threadIdx.x & 31) == 0) {
        atomicAdd(&stats[o], s);
        atomicAdd(&stats[OC + o], ss);
    }
}

// ---------------------------------------------------------------------------
// Kernel 4: normalize + LeakyReLU + max over k. One block per (b, o).
// ---------------------------------------------------------------------------
__global__ __launch_bounds__(256) void ec_final(const float* __restrict__ y,
                                                const int* __restrict__ idxb,
                                                const float* __restrict__ stats,
                                                const float* __restrict__ gamma,
                                                const float* __restrict__ beta,
                                                float* __restrict__ out) {
    __shared__ float y1s[NPTS];
    __shared__ float y2s[NPTS];
    const int b = blockIdx.x >> 6;
    const int o = blockIdx.x & 63;
    const float* y1 = y + (b * 128 + o) * NPTS;
    const float* y2 = y + (b * 128 + OC + o) * NPTS;
    for (int n = threadIdx.x; n < NPTS; n += blockDim.x) {
        y1s[n] = y1[n]; y2s[n] = y2[n];
    }
    __syncthreads();

    const float inv = 1.0f / (float)CNT_ELEMS;
    float mean = stats[o] * inv;
    float var  = stats[OC + o] * inv - mean * mean;
    float scale = gamma[o] * rsqrtf(var + 1e-5f);
    float shift = beta[o] - mean * scale;

    for (int n = threadIdx.x; n < NPTS; n += blockDim.x) {
        float base = y2s[n];
        const int* ir = idxb + (b * NPTS + n) * KNN;
        float best = -3.0e38f;
#pragma unroll
        for (int k = 0; k < KNN; ++k) {
            float v = scale * (y1s[ir[k]] + base) + shift;
            v = (v >= 0.0f) ? v : 0.2f * v;
            best = fmaxf(best, v);
        }
        out[(b * OC + o) * NPTS + n] = best;
    }
}

// ---------------------------------------------------------------------------
extern "C" void kernel_launch(void* const* d_in, const int* in_sizes, int n_in,
                              void* d_out, int out_size, void* d_ws, size_t ws_size,
                              hipStream_t stream) {
    (void)in_sizes; (void)n_in; (void)out_size; (void)ws_size;
    const float* x     = (const float*)d_in[0];
    const float* W0    = (const float*)d_in[1];
    const float* gamma = (const float*)d_in[2];
    const float* beta  = (const float*)d_in[3];
    float* out = (float*)d_out;

    // Workspace layout (~19.4 MB total)
    float* Wc    = (float*)d_ws;                        // 128*32 floats
    float* stats = Wc + 128 * NCH;                      // 128 floats
    int*   idxb  = (int*)(stats + 128);                 // B*N*K ints  (2.62 MB)
    float* y     = (float*)(idxb + BATCH * NPTS * KNN); // B*128*N floats (16.8 MB)

    ec_prep <<<16, 256, 0, stream>>>(W0, Wc, stats);
    ec_knn  <<<BATCH * (NPTS / 128), 128, 0, stream>>>(x, idxb);
    ec_gemm <<<BATCH * (NPTS / 16), 128, 0, stream>>>(x, Wc, y);
    ec_stats<<<BATCH * OC, 256, 0, stream>>>(y, idxb, stats);
    ec_final<<<BATCH * OC, 256, 0, stream>>>(y, idxb, stats, gamma, beta, out);
}